// TGMAPE_31344671326261
// MI455X (gfx1250) — compile-verified
//
#include <hip/hip_runtime.h>
#include <hip/hip_bf16.h>
#include <cstddef>

#define LQ 3
#define DQ 64
#define BQ 8192
#define PQ 20
#define NUQ 400000
#define NVQ 600000
#define NPQ 100000
#define NTQ 1000

typedef __attribute__((ext_vector_type(2))) float v2f;
typedef __attribute__((ext_vector_type(4))) float f4;
typedef __attribute__((ext_vector_type(8))) float v8f;

// -------- CDNA5 WMMA f32 16x16x4 wrapper --------
__device__ __forceinline__ v8f wmma_f32_k4(v8f c, v2f a, v2f b) {
  return __builtin_amdgcn_wmma_f32_16x16x4_f32(
      /*neg_a=*/false, a, /*neg_b=*/false, b,
      /*c_mod=*/(short)0, c, /*reuse_a=*/false, /*reuse_b=*/false);
}

// C[16][16] (cols ncol0..+15) = A_lds[16][K] * W[K][ldw]
// A frag: lane holds A[lane&15][k + 2*(lane>>4) + {0,1}]
// B frag: VGPR0 = W[k+2*hi][ncol0 + (lane&15)], VGPR1 = next K row
__device__ __forceinline__ v8f gemm16(const float* Alds, int lda,
                                      const float* __restrict__ W, int ldw,
                                      int K, int ncol0, int lane) {
  const int r  = lane & 15;
  const int hi = lane >> 4;
  v8f c = {0.f, 0.f, 0.f, 0.f, 0.f, 0.f, 0.f, 0.f};
  for (int k = 0; k < K; k += 4) {
    const int ka = k + 2 * hi;
    v2f a, b;
    a.x = Alds[r * lda + ka];
    a.y = Alds[r * lda + ka + 1];
    b.x = W[(size_t)ka * ldw + ncol0 + r];
    b.y = W[(size_t)(ka + 1) * ldw + ncol0 + r];
    c = wmma_f32_k4(c, a, b);
  }
  return c;
}

// C/D layout: c[i] -> row (i + 8*hi), col ncol0 + (lane&15)
__device__ __forceinline__ void frag_store(float* out, int ldo, int ncol0, v8f c,
                                           const float* __restrict__ bias,
                                           bool lrelu, int lane) {
  const int n = lane & 15, hi = lane >> 4;
  const float bv = bias ? bias[ncol0 + n] : 0.f;
#pragma unroll
  for (int i = 0; i < 8; i++) {
    float v = c[i] + bv;
    if (lrelu) v = (v > 0.f) ? v : 0.01f * v;
    out[(i + 8 * hi) * ldo + ncol0 + n] = v;
  }
}

__device__ __forceinline__ float wave_sum(float v) {
#pragma unroll
  for (int o = 16; o > 0; o >>= 1) v += __shfl_xor(v, o, 32);
  return v;
}
__device__ __forceinline__ float wave_max(float v) {
#pragma unroll
  for (int o = 16; o > 0; o >>= 1) v = fmaxf(v, __shfl_xor(v, o, 32));
  return v;
}
__device__ __forceinline__ float softplusf(float v) {  // log(1+e^v), stable
  return fmaxf(v, 0.f) + log1pf(expf(-fabsf(v)));
}
__device__ __forceinline__ float logsigf(float v) {    // -softplus(-v)
  return fminf(v, 0.f) - log1pf(expf(-fabsf(v)));
}
__device__ __forceinline__ float dot4(f4 a, f4 b) {
  return a.x * b.x + a.y * b.y + a.z * b.z + a.w * b.w;
}

// =====================================================================
// feat_kernel: one wave per 16 batch rows. Builds DNN input x[B][256]:
// [user_e | video_agg | pub_agg | domain_e]
// Publisher embeddings streamed via async-to-LDS (double buffered).
// =====================================================================
__global__ __launch_bounds__(32) void feat_kernel(
    const int* __restrict__ user, const int* __restrict__ video,
    const int* __restrict__ publishers, const int* __restrict__ tags,
    const int* __restrict__ domain_id,
    const float* __restrict__ user_tabs, const float* __restrict__ video_tabs,
    const float* __restrict__ pub_tabs, const float* __restrict__ tag_tabs,
    const float* __restrict__ domain_tab,
    const float* __restrict__ key_W, const float* __restrict__ key_b,
    float* __restrict__ x) {
  __shared__ float As[16][128];     // concat(video_feat, tag_e)
  __shared__ float keyt[16][64];    // key_emb tile
  __shared__ f4 pb[2][PQ * 16];     // double-buffered pub embeddings (1 row)
  __shared__ f4 pagg4[16 * 16];     // pub_agg accumulator over levels
  const int lane = threadIdx.x;
  const int r0 = blockIdx.x * 16;

  // issue one row's 20x64f pub block as async b128 loads into dst
  auto issue_row = [&](int l, f4* dst, int idxlane) {
    for (int t = lane; t < PQ * 16; t += 32) {
      int p = t >> 4, q = t & 15;
      int pid = __shfl(idxlane, p, 32);
      const float* src = pub_tabs + ((size_t)l * NPQ + pid) * DQ + q * 4;
      unsigned lo = (unsigned)(size_t)(dst + (size_t)p * 16 + q);
      unsigned long long ga = (unsigned long long)(size_t)src;
      asm volatile("global_load_async_to_lds_b128 %0, %1, off"
                   :: "v"(lo), "v"(ga) : "memory");
    }
  };

  for (int l = 0; l < LQ; l++) {
    // stage A = [video_tabs[l,vid] | tag_tabs[l,tag]]  (b128 chunks)
    f4* As4 = (f4*)&As[0][0];
    for (int t = lane; t < 16 * 32; t += 32) {
      int m = t >> 5, q = t & 31, b = r0 + m;
      f4 v;
      if (q < 16)
        v = ((const f4*)(video_tabs + ((size_t)l * NVQ + video[b]) * DQ))[q];
      else
        v = ((const f4*)(tag_tabs + ((size_t)l * NTQ + tags[b * LQ + l]) * DQ))[q - 16];
      As4[t] = v;
    }
    __syncthreads();
    // key_emb = A @ key_W + key_b
    for (int nf = 0; nf < 4; nf++) {
      v8f cf = gemm16(&As[0][0], 128, key_W, 64, 128, nf * 16, lane);
      frag_store(&keyt[0][0], 64, nf * 16, cf, key_b, false, lane);
    }
    __syncthreads();

    // ---- pipelined per-row attention over 20 publishers ----
    int curIdx = (lane < PQ) ? publishers[(size_t)r0 * PQ + lane] : 0;
    issue_row(l, pb[0], curIdx);
    for (int m = 0; m < 16; m++) {
      int nxtIdx = 0;
      if (m < 15) {
        nxtIdx = (lane < PQ) ? publishers[(size_t)(r0 + m + 1) * PQ + lane] : 0;
        issue_row(l, pb[(m + 1) & 1], nxtIdx);
        asm volatile("s_wait_asynccnt 0xa" ::: "memory");  // next row may stay in flight
      } else {
        asm volatile("s_wait_asynccnt 0x0" ::: "memory");
      }
      __syncthreads();
      const f4* pub = pb[m & 1];
      // scores (20 lanes, f4 dots against key_emb row)
      float s = -1e30f;  // inert filler for lanes >= P
      if (lane < PQ) {
        const f4* kt4 = (const f4*)&keyt[m][0];
        const f4* pr4 = pub + (size_t)lane * 16;
        float acc = 0.f;
#pragma unroll
        for (int q = 0; q < 16; q++) acc += dot4(kt4[q], pr4[q]);
        s = (curIdx > 0) ? acc : -1e9f;  // reference mask value
      }
      float mx = wave_max(s);
      float e = (lane < PQ) ? expf(s - mx) : 0.f;
      float sm = wave_sum(e);
      float attn = e / sm;
      // weighted sum: lane owns f4 chunk (lane&15), halves split p-range
      int chunk = lane & 15, half = lane >> 4;
      f4 acc4 = {0.f, 0.f, 0.f, 0.f};
      for (int q = half * 10; q < half * 10 + 10; q++) {
        float w = __shfl(attn, q, 32);
        acc4 += pub[(size_t)q * 16 + chunk] * w;
      }
      acc4.x += __shfl_xor(acc4.x, 16, 32);
      acc4.y += __shfl_xor(acc4.y, 16, 32);
      acc4.z += __shfl_xor(acc4.z, 16, 32);
      acc4.w += __shfl_xor(acc4.w, 16, 32);
      if (lane < 16)
        pagg4[m * 16 + chunk] = (l == 0) ? acc4 : (pagg4[m * 16 + chunk] + acc4);
      curIdx = nxtIdx;
      __syncthreads();
    }
    __syncthreads();
  }
  // emit x = [user_e | video_agg | pub_agg/3 | domain_e]  (b128 stores)
  for (int t = lane; t < 16 * 16; t += 32) {
    int m = t >> 4, q = t & 15, b = r0 + m;
    int u = user[b], vid = video[b];
    f4 ue = {0.f, 0.f, 0.f, 0.f}, va = {0.f, 0.f, 0.f, 0.f};
    for (int l = 0; l < LQ; l++) {
      ue += ((const f4*)(user_tabs + ((size_t)l * NUQ + u) * DQ))[q];
      va += ((const f4*)(video_tabs + ((size_t)l * NVQ + vid) * DQ))[q];
    }
    f4* xr = (f4*)(x + (size_t)b * 256);
    xr[q]      = ue * (1.f / 3.f);
    xr[16 + q] = va * (1.f / 3.f);
    xr[32 + q] = pagg4[m * 16 + q] * (1.f / 3.f);
    xr[48 + q] = ((const f4*)(domain_tab + (size_t)domain_id[b] * DQ))[q];
  }
}

// =====================================================================
// loss_kernel: blockIdx encodes (tile, lossType in {par,bro}, i in {1,2}).
// =====================================================================
__global__ __launch_bounds__(32) void loss_kernel(
    const int* __restrict__ video,
    const int* __restrict__ neg_par_video,
    const int* __restrict__ pos_bro_video,
    const int* __restrict__ neg_bro_video,
    const float* __restrict__ video_tabs,
    const float* __restrict__ par_W1, const float* __restrict__ par_b1,
    const float* __restrict__ par_W2, const float* __restrict__ par_b2,
    const float* __restrict__ bro_W1, const float* __restrict__ bro_b1,
    const float* __restrict__ bro_W2, const float* __restrict__ bro_b2,
    float* __restrict__ lossPart) {
  __shared__ float As[16][128];
  __shared__ float h[16][64];
  const int lane = threadIdx.x;
  const int combo = blockIdx.x & 3;
  const int r0 = (blockIdx.x >> 2) * 16;
  const int isBro = combo >> 1;
  const int i = 1 + (combo & 1);
  const float* W1 = isBro ? bro_W1 : par_W1;
  const float* b1 = isBro ? bro_b1 : par_b1;
  const float* W2 = isBro ? bro_W2 : par_W2;
  const float  b2 = (isBro ? bro_b2 : par_b2)[0];

  float sc0 = 0.f, sc1 = 0.f;
  f4* As4 = (f4*)&As[0][0];
  for (int side = 0; side < 2; side++) {
    for (int t = lane; t < 16 * 32; t += 32) {
      int m = t >> 5, q = t & 31, b = r0 + m;
      int lvl, idx;
      if (!isBro) {
        // par: concat([pos_par|neg_par], pos_video)
        if (q < 16) { lvl = i - 1; idx = (side == 0) ? video[b] : neg_par_video[b * LQ + (i - 1)]; }
        else        { lvl = i;     idx = video[b]; }
      } else {
        // bro: concat(v, [neg_bro|pos_bro])  (reference naming swap preserved)
        if (q < 16) { lvl = i; idx = video[b]; }
        else        { lvl = i; idx = (side == 0) ? neg_bro_video[b * LQ + i] : pos_bro_video[b * LQ + i]; }
      }
      As4[t] = ((const f4*)(video_tabs + ((size_t)lvl * NVQ + idx) * DQ))[q & 15];
    }
    __syncthreads();
    for (int nf = 0; nf < 4; nf++) {
      v8f cf = gemm16(&As[0][0], 128, W1, 64, 128, nf * 16, lane);
      frag_store(&h[0][0], 64, nf * 16, cf, b1, true, lane);
    }
    __syncthreads();
    // per-row W2 dot using full wave: lane halves split d-range
    int row = lane & 15, half = lane >> 4;
    const f4* h4 = (const f4*)&h[row][0];
    const f4* w24 = (const f4*)W2;
    float sc = 0.f;
#pragma unroll
    for (int q = 0; q < 8; q++) sc += dot4(h4[half * 8 + q], w24[half * 8 + q]);
    sc += __shfl_xor(sc, 16, 32);
    sc += b2;
    if (side == 0) sc0 = sc; else sc1 = sc;
    __syncthreads();
  }
  float ls = (lane < 16) ? logsigf(sc0 - sc1) : 0.f;
  float tot = wave_sum(ls);
  if (lane == 0) lossPart[blockIdx.x] = tot;
}

// =====================================================================
// dnn_kernel: x[16][256] -> W0 -> lrelu -> W1 -> lrelu -> W2 -> sigmoid
// =====================================================================
__global__ __launch_bounds__(32) void dnn_kernel(
    const float* __restrict__ x, const int* __restrict__ long_view,
    const float* __restrict__ W0, const float* __restrict__ b0,
    const float* __restrict__ W1, const float* __restrict__ b1,
    const float* __restrict__ W2, const float* __restrict__ b2,
    float* __restrict__ task_out, float* __restrict__ bprPart) {
  __shared__ float A0[16][256];
  __shared__ float h1[16][128];
  __shared__ float h2[16][64];
  const int lane = threadIdx.x;
  const int r0 = blockIdx.x * 16;
  f4* A04 = (f4*)&A0[0][0];
  const f4* x4 = (const f4*)x;
  for (int t = lane; t < 16 * 64; t += 32) {
    int m = t >> 6, q = t & 63;
    A04[t] = x4[(size_t)(r0 + m) * 64 + q];
  }
  __syncthreads();
  for (int nf = 0; nf < 8; nf++) {
    v8f cf = gemm16(&A0[0][0], 256, W0, 128, 256, nf * 16, lane);
    frag_store(&h1[0][0], 128, nf * 16, cf, b0, true, lane);
  }
  __syncthreads();
  for (int nf = 0; nf < 4; nf++) {
    v8f cf = gemm16(&h1[0][0], 128, W1, 64, 128, nf * 16, lane);
    frag_store(&h2[0][0], 64, nf * 16, cf, b1, true, lane);
  }
  __syncthreads();
  // per-row W2 dot with full wave
  int row = lane & 15, half = lane >> 4;
  const f4* h4 = (const f4*)&h2[row][0];
  const f4* w24 = (const f4*)W2;
  float logit = 0.f;
#pragma unroll
  for (int q = 0; q < 8; q++) logit += dot4(h4[half * 8 + q], w24[half * 8 + q]);
  logit += __shfl_xor(logit, 16, 32);
  logit += b2[0];
  float term = 0.f;
  if (lane < 16) {
    task_out[r0 + row] = 1.f / (1.f + expf(-logit));
    float y = (float)long_view[r0 + row];
    term = y * softplusf(-logit) + (1.f - y) * softplusf(logit);
  }
  float tot = wave_sum(term);
  if (lane == 0) bprPart[blockIdx.x] = tot;
}

// =====================================================================
// final_kernel: deterministic fixed-order reduction of partials
// =====================================================================
__global__ __launch_bounds__(32) void final_kernel(
    const float* __restrict__ bprPart, const float* __restrict__ lossPart,
    float* __restrict__ out) {
  const int lane = threadIdx.x;
  float accB = 0.f;
  for (int k = lane; k < BQ / 16; k += 32) accB += bprPart[k];
  float accI = 0.f, accT = 0.f;
  for (int k = lane; k < (BQ / 16) * 4; k += 32) {
    float v = lossPart[k];
    if (((k >> 1) & 1) == 0) accI += v; else accT += v;
  }
  accB = wave_sum(accB);
  accI = wave_sum(accI);
  accT = wave_sum(accT);
  if (lane == 0) {
    out[0] = accB / (float)BQ;                              // bpr
    out[1] = -0.1f * accI / ((float)BQ * (float)(LQ - 1));  // inter
    out[2] = -0.1f * accT / ((float)BQ * (float)(LQ - 1));  // intra
  }
}

extern "C" void kernel_launch(void* const* d_in, const int* in_sizes, int n_in,
                              void* d_out, int out_size, void* d_ws, size_t ws_size,
                              hipStream_t stream) {
  (void)in_sizes; (void)n_in; (void)out_size; (void)ws_size;
  const int* user       = (const int*)d_in[0];
  const int* video      = (const int*)d_in[1];
  const int* publishers = (const int*)d_in[2];
  const int* tags       = (const int*)d_in[3];
  const int* domain_id  = (const int*)d_in[4];
  const int* neg_par    = (const int*)d_in[5];
  const int* pos_bro    = (const int*)d_in[6];
  const int* neg_bro    = (const int*)d_in[7];
  const int* long_view  = (const int*)d_in[8];
  const float* user_tabs  = (const float*)d_in[9];
  const float* video_tabs = (const float*)d_in[10];
  const float* pub_tabs   = (const float*)d_in[11];
  const float* tag_tabs   = (const float*)d_in[12];
  const float* domain_tab = (const float*)d_in[13];
  const float* key_W  = (const float*)d_in[14];
  const float* key_b  = (const float*)d_in[15];
  const float* par_W1 = (const float*)d_in[16];
  const float* par_b1 = (const float*)d_in[17];
  const float* par_W2 = (const float*)d_in[18];
  const float* par_b2 = (const float*)d_in[19];
  const float* bro_W1 = (const float*)d_in[20];
  const float* bro_b1 = (const float*)d_in[21];
  const float* bro_W2 = (const float*)d_in[22];
  const float* bro_b2 = (const float*)d_in[23];
  const float* dnn_W0 = (const float*)d_in[24];
  const float* dnn_b0 = (const float*)d_in[25];
  const float* dnn_W1 = (const float*)d_in[26];
  const float* dnn_b1 = (const float*)d_in[27];
  const float* dnn_W2 = (const float*)d_in[28];
  const float* dnn_b2 = (const float*)d_in[29];

  float* out = (float*)d_out;  // [bpr, inter, intra, task_out(8192)]
  char* ws = (char*)d_ws;
  float* x        = (float*)ws;                            // B*256 f32 = 8 MB
  float* bprPart  = (float*)(ws + (size_t)BQ * 256 * 4);   // 512 f32
  float* lossPart = bprPart + (BQ / 16);                   // 2048 f32

  feat_kernel<<<BQ / 16, 32, 0, stream>>>(
      user, video, publishers, tags, domain_id,
      user_tabs, video_tabs, pub_tabs, tag_tabs, domain_tab, key_W, key_b, x);
  loss_kernel<<<(BQ / 16) * 4, 32, 0, stream>>>(
      video, neg_par, pos_bro, neg_bro, video_tabs,
      par_W1, par_b1, par_W2, par_b2, bro_W1, bro_b1, bro_W2, bro_b2, lossPart);
  dnn_kernel<<<BQ / 16, 32, 0, stream>>>(
      x, long_view, dnn_W0, dnn_b0, dnn_W1, dnn_b1, dnn_W2, dnn_b2,
      out + 3, bprPart);
  final_kernel<<<1, 32, 0, stream>>>(bprPart, lossPart, out);
}